// VectorQuantizer_17532056502308
// MI455X (gfx1250) — compile-verified
//
#include <hip/hip_runtime.h>

typedef float v2f __attribute__((ext_vector_type(2)));
typedef float v8f __attribute__((ext_vector_type(8)));
typedef unsigned int u32x4 __attribute__((ext_vector_type(4)));
typedef int i32x4 __attribute__((ext_vector_type(4)));
typedef int i32x8 __attribute__((ext_vector_type(8)));

#define NROWS   32768      // B*D*H*W = 8*16*16*16
#define NCODES  2048
#define CDIM    256
#define SPATIAL 4096       // D*H*W
#define M_TILE  64
#define N_CHUNK 32
#define NCHUNKS (NCODES / N_CHUNK)
#define ZSTR    260        // 256 + 4 pad -> conflict-free fragment loads
#define ESTR    260        // TDM reproduces this via pad_interval/pad_amount
#define SSTR    36
#define NQ      8388608    // 8*256*16*16*16
#define LOSS_BLOCKS 8192

// ---------------- Kernel 1: ||e_k||^2 ----------------
__global__ __launch_bounds__(256) void vq_prep(const float* __restrict__ emb,
                                               float* __restrict__ e2) {
    int code = blockIdx.x * blockDim.x + threadIdx.x;
    if (code < NCODES) {
        const float* row = emb + (size_t)code * CDIM;
        float s = 0.f;
#pragma unroll 8
        for (int c = 0; c < CDIM; ++c) s += row[c] * row[c];
        e2[code] = s;
    }
}

// TDM: DMA one [N_CHUNK x 256] f32 emb tile Global->LDS with +4-DWORD row padding.
// D# per CDNA5 ISA: group0 {count,lds,global,type=2},
// group1 {data_size=4B, pad(4DW per 256DW), dims/strides}, groups2/3 zero (2D).
__device__ __forceinline__ void tdm_load_emb_chunk(const float* gsrc, unsigned lds_byte_off) {
    unsigned long long ga = (unsigned long long)(const void*)gsrc;
    u32x4 g0;
    g0[0] = 1u;                                                   // count=1 (valid user D#)
    g0[1] = lds_byte_off;                                         // lds_addr (bytes)
    g0[2] = (unsigned)(ga & 0xFFFFFFFFu);                         // global_addr[31:0]
    g0[3] = (unsigned)((ga >> 32) & 0x01FFFFFFu) | (2u << 30);    // addr[56:32] | type=2

    i32x8 g1;
    g1[0] = (int)((2u << 16) | (1u << 20) | (7u << 22) | (3u << 25)); // 4B elems, pad_en, 256DW interval, 4DW pad
    g1[1] = (int)(256u << 16);        // tensor_dim0 = 256 (low 16 in bits 63:48)
    g1[2] = (int)(2048u << 16);       // tensor_dim0 hi=0 | tensor_dim1 = 2048
    g1[3] = (int)(256u << 16);        // tensor_dim1 hi=0 | tile_dim0 = 256
    g1[4] = (int)(N_CHUNK);           // tile_dim1 = 32 | tile_dim2 = 0
    g1[5] = (int)(256);               // tensor_dim0_stride = 256 (lo32)
    g1[6] = 0;                        // stride0 hi | tensor_dim1_stride lo
    g1[7] = 0;
    i32x4 gz = {0, 0, 0, 0};
#if defined(__clang_major__) && (__clang_major__ >= 23)
    i32x8 gz8 = {0, 0, 0, 0, 0, 0, 0, 0};
    __builtin_amdgcn_tensor_load_to_lds(g0, g1, gz, gz, gz8, 0);
#else
    __builtin_amdgcn_tensor_load_to_lds(g0, g1, gz, gz, 0);
#endif
}

// ---------------- Kernel 2: WMMA f32 score GEMM + TDM double-buffer + argmin ----------------
__global__ __launch_bounds__(256) void vq_argmin(const float* __restrict__ z,
                                                 const float* __restrict__ emb,
                                                 const float* __restrict__ e2,
                                                 int* __restrict__ out_idx) {
    __shared__ float z_lds[M_TILE * ZSTR];            // holds -2*z tile
    __shared__ float e_lds[2 * N_CHUNK * ESTR];       // TDM ping-pong buffers
    __shared__ float s_lds[M_TILE * SSTR];

    const int tid  = threadIdx.x;
    const int wave = tid >> 5;
    const int row0 = blockIdx.x * M_TILE;

    // Prime the pipeline: wave 0 issues TDM for chunk 0 into buffer 0.
    if (wave == 0) {
        tdm_load_emb_chunk(emb, (unsigned)(size_t)(void*)&e_lds[0]);
    }

    // Stage -2 * z_flat[row0 .. row0+63][0..255] into LDS (overlaps with the DMA).
    // z_flat[n][c] = z[(n>>12)*256*4096 + c*4096 + (n&4095)]  (channels-last permute)
    {
        const int r  = tid & 63;
        const int c0 = tid >> 6;       // 0..3
        const int n  = row0 + r;
        const float* zb = z + ((size_t)(n >> 12) * CDIM) * SPATIAL + (n & (SPATIAL - 1));
        for (int c = c0; c < CDIM; c += 4)
            z_lds[r * ZSTR + c] = -2.0f * zb[(size_t)c * SPATIAL];
    }

    const int lane  = tid & 31;
    const int wm    = wave & 3;        // 4 waves along M
    const int wn    = wave >> 2;       // 2 waves along N
    const int l16   = lane & 15;
    const int half  = lane >> 4;       // 0: K pair {0,1}; 1: K pair {2,3}
    const int mbase = wm * 16;
    const int nbase = wn * 16;

    float run_min = 3.4e38f;
    int   run_idx = 0;

    __syncthreads();                    // z tile visible to all waves

    for (int chunk = 0; chunk < NCHUNKS; ++chunk) {
        const int cur   = chunk & 1;
        const int code0 = chunk * N_CHUNK;

        if (wave == 0) __builtin_amdgcn_s_wait_tensorcnt(0);  // chunk's DMA landed
        __syncthreads();                                      // publish e_lds[cur]

        // Prefetch next chunk into the other buffer while we compute this one.
        if (wave == 0 && chunk + 1 < NCHUNKS) {
            tdm_load_emb_chunk(emb + (size_t)(code0 + N_CHUNK) * CDIM,
                               (unsigned)(size_t)(void*)&e_lds[(1 - cur) * N_CHUNK * ESTR]);
        }

        // Accumulator initialized with ||e||^2: C/D layout has N = lane%16 in every VGPR.
        const float e2v = e2[code0 + nbase + l16];
        v8f acc = { e2v, e2v, e2v, e2v, e2v, e2v, e2v, e2v };

        // A frag (16x4, f32): lane<16 -> (M=l16, K={k,k+1}); lane>=16 -> K={k+2,k+3}
        // B frag (4x16, f32): lane<16 -> (N=l16, K={k,k+1}); lane>=16 -> K={k+2,k+3}
        const float* ap = z_lds + (mbase + l16) * ZSTR + half * 2;
        const float* bp = e_lds + (size_t)cur * N_CHUNK * ESTR + (nbase + l16) * ESTR + half * 2;
#pragma unroll 8
        for (int k = 0; k < CDIM; k += 4) {
            v2f a = *(const v2f*)(ap + k);
            v2f b = *(const v2f*)(bp + k);
            acc = __builtin_amdgcn_wmma_f32_16x16x4_f32(
                false, a, false, b, (short)0, acc, false, false);
        }

        // Scatter scores s = ||e||^2 - 2 z.e to LDS.
#pragma unroll
        for (int r = 0; r < 8; ++r) {
            const int rr = mbase + r + half * 8;
            s_lds[rr * SSTR + nbase + l16] = acc[r];
        }
        __syncthreads();

        // 64 threads own one row each; running argmin, first-hit tie-break like jnp.argmin.
        if (tid < M_TILE) {
            const float* srow = s_lds + tid * SSTR;
            for (int j = 0; j < N_CHUNK; ++j) {
                const float v = srow[j];
                if (v < run_min) { run_min = v; run_idx = code0 + j; }
            }
        }
        __syncthreads();   // protect s_lds (and retire e_lds[cur] reads) before next chunk
    }

    if (tid < M_TILE) out_idx[row0 + tid] = run_idx;
}

// ---------------- Kernel 3: gather z_q (with faithful reshape bug) + loss partials ----------------
__global__ __launch_bounds__(256) void vq_gather_loss(const float* __restrict__ z,
                                                      const float* __restrict__ emb,
                                                      const int* __restrict__ idx,
                                                      float* __restrict__ out_q,
                                                      float* __restrict__ partial) {
    __shared__ float red[256];
    const size_t i = ((size_t)blockIdx.x * blockDim.x + threadIdx.x) * 4;
    const int n = (int)(i >> 8);      // row in permuted flatten
    const int c = (int)(i & 255);
    const int code = idx[n];

    const float4 q  = *(const float4*)(emb + (size_t)code * CDIM + c);
    const float4 zv = *(const float4*)(z + i);        // ORIGINAL layout (the faithful bug)
    *(float4*)(out_q + i) = q;                        // z_q_st == z_q numerically

    const float d0 = q.x - zv.x, d1 = q.y - zv.y, d2 = q.z - zv.z, d3 = q.w - zv.w;
    red[threadIdx.x] = d0*d0 + d1*d1 + d2*d2 + d3*d3;
    __syncthreads();
#pragma unroll
    for (int off = 128; off > 0; off >>= 1) {
        if (threadIdx.x < off) red[threadIdx.x] += red[threadIdx.x + off];
        __syncthreads();
    }
    if (threadIdx.x == 0) partial[blockIdx.x] = red[0];
}

// ---------------- Kernel 4: deterministic final reduction ----------------
__global__ __launch_bounds__(256) void vq_loss_final(const float* __restrict__ partial,
                                                     float* __restrict__ loss) {
    __shared__ float red[256];
    float s = 0.f;
    for (int i = threadIdx.x; i < LOSS_BLOCKS; i += 256) s += partial[i];
    red[threadIdx.x] = s;
    __syncthreads();
#pragma unroll
    for (int off = 128; off > 0; off >>= 1) {
        if (threadIdx.x < off) red[threadIdx.x] += red[threadIdx.x + off];
        __syncthreads();
    }
    if (threadIdx.x == 0) *loss = red[0] * (1.25f / (float)NQ);   // (1+BETA)*mean
}

extern "C" void kernel_launch(void* const* d_in, const int* in_sizes, int n_in,
                              void* d_out, int out_size, void* d_ws, size_t ws_size,
                              hipStream_t stream) {
    const float* z   = (const float*)d_in[0];   // (8,256,16,16,16) fp32
    const float* emb = (const float*)d_in[1];   // (2048,256) fp32

    float* out_q    = (float*)d_out;            // [8388608] z_q_st
    float* out_loss = out_q + NQ;               // [1] vq_loss
    int*   out_idx  = (int*)(out_q + NQ + 1);   // [32768] int32 idx

    float* e2      = (float*)d_ws;              // [2048]
    float* partial = e2 + NCODES;               // [8192]

    vq_prep       <<<NCODES / 256, 256, 0, stream>>>(emb, e2);
    vq_argmin     <<<NROWS / M_TILE, 256, 0, stream>>>(z, emb, e2, out_idx);
    vq_gather_loss<<<LOSS_BLOCKS, 256, 0, stream>>>(z, emb, out_idx, out_q, partial);
    vq_loss_final <<<1, 256, 0, stream>>>(partial, out_loss);
}